// PCENLayer_30468497998075
// MI455X (gfx1250) — compile-verified
//
#include <hip/hip_runtime.h>

// PCEN for x[B=64][T=2048][F=160][C=1], f32.
// EMA scan recast as truncated 32-tap causal convolution (s=0.03 -> s^16 ~ 4e-25,
// far below f32 eps), computed with V_WMMA_F32_16X16X4_F32:
//   D(16t x 16f) = W_halo(16x16) x X_prev(16t x 16f) + W_tri(16x16) x X_cur
// One wave per (batch, 16-channel f-tile, 128-timestep chunk); X_cur of tile i is
// reused as X_prev of tile i+1 in registers.

typedef __attribute__((ext_vector_type(2))) float v2f;
typedef __attribute__((ext_vector_type(8))) float v8f;

namespace {
constexpr int kT          = 2048;
constexpr int kF          = 160;
constexpr int kTile       = 16;                        // timesteps per WMMA tile
constexpr int kChunkTiles = 8;                         // 128 timesteps per wave
constexpr int kChunks     = kT / (kTile * kChunkTiles);// 16
constexpr int kFTiles     = kF / 16;                   // 10
constexpr float kEps      = 1e-6f;
}

// exp2 via hardware v_exp_f32 (device __exp2f is missing on this toolchain).
__device__ __forceinline__ float hw_exp2(float v) {
  return __builtin_amdgcn_exp2f(v);
}

__global__ __launch_bounds__(256) void pcen_wmma_kernel(
    const float* __restrict__ x,
    const float* __restrict__ p_alpha,
    const float* __restrict__ p_smooth,
    const float* __restrict__ p_delta,
    const float* __restrict__ p_root,
    float* __restrict__ out,
    int n_waves)
{
  const int wid  = (int)((blockIdx.x * blockDim.x + threadIdx.x) >> 5);
  const int lane = (int)(threadIdx.x & 31);
  if (wid >= n_waves) return;  // wave-uniform: EXEC stays all-ones for WMMA

  const float alpha = p_alpha[0];
  const float s     = p_smooth[0];
  const float delta = p_delta[0];
  const float root  = p_root[0];
  const float oms   = 1.0f - s;
  const float droot = hw_exp2(root * __log2f(delta));   // delta^root

  const int chunk = wid % kChunks;
  const int ft    = (wid / kChunks) % kFTiles;
  const int b     = wid / (kChunks * kFTiles);

  const int m  = lane & 15;  // A row (time-in-tile) == B/D column (f-in-tile)
  const int hi = lane >> 4;  // half-wave select: K+2 for A/B, M+8 for C/D

  // Column base: element for timestep t of channel (ft*16 + m) is xp[t*kF].
  const size_t colbase = ((size_t)b * kT) * kF + (size_t)ft * 16 + m;
  const float* xp = x   + colbase;
  float*       op = out + colbase;

  // ---- A operands (ISA f32 16x4 A layout: lane m holds K = 4ks+2hi, +1) ----
  // W_tri[i][j]  = (j<=i) ? (1-s)*s^(i-j) : 0      (current tile, causal)
  // W_halo[i][j] = (1-s)*s^(i-j+16)                (previous-tile halo)
  v2f A1[4], A2[4];
#pragma unroll
  for (int ks = 0; ks < 4; ++ks) {
    const int k0 = 4 * ks + 2 * hi;
    A2[ks][0] = (k0     <= m) ? oms * __powf(s, (float)(m - k0))     : 0.0f;
    A2[ks][1] = (k0 + 1 <= m) ? oms * __powf(s, (float)(m - k0 - 1)) : 0.0f;
    A1[ks][0] = oms * __powf(s, (float)(m - k0 + 16));
    A1[ks][1] = oms * __powf(s, (float)(m - k0 + 15));
  }

  const int t0c = chunk * (kTile * kChunkTiles);

  // ---- Halo tile (B-operand layout: lane n holds rows K = 4ks+2hi, +1) ----
  v2f Bprev[4], Bcur[4];
  if (t0c == 0) {
#pragma unroll
    for (int ks = 0; ks < 4; ++ks) { Bprev[ks][0] = 0.0f; Bprev[ks][1] = 0.0f; }
  } else {
#pragma unroll
    for (int ks = 0; ks < 4; ++ks) {
      const int t = (t0c - kTile) + 4 * ks + 2 * hi;
      Bprev[ks][0] = xp[(size_t)t * kF];
      Bprev[ks][1] = xp[(size_t)(t + 1) * kF];
    }
  }

  for (int tt = 0; tt < kChunkTiles; ++tt) {
    const int t0 = t0c + tt * kTile;

#pragma unroll
    for (int ks = 0; ks < 4; ++ks) {
      const int t = t0 + 4 * ks + 2 * hi;
      Bcur[ks][0] = xp[(size_t)t * kF];
      Bcur[ks][1] = xp[(size_t)(t + 1) * kF];
    }

    // D = W_halo x X_prev + W_tri x X_cur   (8 chained f32 WMMAs, K=4 each)
    v8f acc = {0.f, 0.f, 0.f, 0.f, 0.f, 0.f, 0.f, 0.f};
#pragma unroll
    for (int ks = 0; ks < 4; ++ks)
      acc = __builtin_amdgcn_wmma_f32_16x16x4_f32(
          false, A1[ks], false, Bprev[ks], (short)0, acc, false, false);
#pragma unroll
    for (int ks = 0; ks < 4; ++ks)
      acc = __builtin_amdgcn_wmma_f32_16x16x4_f32(
          false, A2[ks], false, Bcur[ks], (short)0, acc, false, false);

    // ---- Elementwise PCEN in C/D layout: VGPR r -> t = t0 + r + 8*hi ----
    // x re-read lands in L2 (84 MB tensor < 192 MB L2), HBM traffic stays ~2N.
#pragma unroll
    for (int r = 0; r < 8; ++r) {
      const int t = t0 + r + 8 * hi;
      const float xv    = xp[(size_t)t * kF];
      const float mm    = acc[r] + kEps;
      const float ratio = xv * hw_exp2(-alpha * __log2f(mm));
      op[(size_t)t * kF] = hw_exp2(root * __log2f(ratio + delta)) - droot;
    }

#pragma unroll
    for (int ks = 0; ks < 4; ++ks) Bprev[ks] = Bcur[ks];
  }
}

extern "C" void kernel_launch(void* const* d_in, const int* in_sizes, int n_in,
                              void* d_out, int out_size, void* d_ws, size_t ws_size,
                              hipStream_t stream) {
  const float* x      = (const float*)d_in[0];
  const float* alpha  = (const float*)d_in[1];
  const float* smooth = (const float*)d_in[2];
  const float* delta  = (const float*)d_in[3];
  const float* root   = (const float*)d_in[4];
  float* out = (float*)d_out;

  const int B       = in_sizes[0] / (kT * kF);           // 64
  const int n_waves = B * kFTiles * kChunks;             // 10240 independent waves
  const int waves_per_block = 8;                         // 256 threads
  const int blocks = (n_waves + waves_per_block - 1) / waves_per_block;

  pcen_wmma_kernel<<<blocks, waves_per_block * 32, 0, stream>>>(
      x, alpha, smooth, delta, root, out, n_waves);
}